// MMD_loss_12378095747598
// MI455X (gfx1250) — compile-verified
//
#include <hip/hip_runtime.h>

typedef __attribute__((ext_vector_type(2))) float v2f;
typedef __attribute__((ext_vector_type(8))) float v8f;

#define NROWS 8192
#define NS    4096
#define DIMS  64
#define NT    512      // 8192/16 tiles per side
#define HALF_T 256     // tiles covering the source half

// ws layout (float indices):
//   [0, 8192)    sq[r] = ||x_r||^2
//   [8192, 8256) colsum[64]
//   [8256]       sumsq (sum of sq[r])
//   [8257]       cExp = 1/(bw*16)
//   [8260..8261] double accumulator (byte offset 33040, 8-byte aligned)
#define WS_SQ     0
#define WS_COL    8192
#define WS_SUMSQ  8256
#define WS_CEXP   8257
#define WS_ACC    8260

__global__ void mmd_init(float* __restrict__ ws) {
    int t = threadIdx.x;
    if (t < 64) ws[WS_COL + t] = 0.f;
    if (t == 64) ws[WS_SUMSQ] = 0.f;
    if (t == 65) *(double*)(ws + WS_ACC) = 0.0;
}

// One thread per row: sq[r], block-local column sums + sumsq, then global atomics.
__global__ void __launch_bounds__(256)
mmd_rowstats(const float* __restrict__ src, const float* __restrict__ tgt,
             float* __restrict__ ws) {
    __shared__ float scol[64];
    __shared__ float ssq;
    const int t = threadIdx.x;
    if (t < 64) scol[t] = 0.f;
    if (t == 64) ssq = 0.f;
    __syncthreads();

    const int r = blockIdx.x * 256 + t;
    const float* row = (r < NS) ? (src + (size_t)r * DIMS)
                                : (tgt + (size_t)(r - NS) * DIMS);
    const float4* row4 = (const float4*)row;

    float sq = 0.f;
#pragma unroll
    for (int i = 0; i < 16; ++i) {
        float4 v = row4[i];
        sq += v.x * v.x + v.y * v.y + v.z * v.z + v.w * v.w;
        atomicAdd(&scol[4 * i + 0], v.x);
        atomicAdd(&scol[4 * i + 1], v.y);
        atomicAdd(&scol[4 * i + 2], v.z);
        atomicAdd(&scol[4 * i + 3], v.w);
    }
    ws[WS_SQ + r] = sq;
    atomicAdd(&ssq, sq);
    __syncthreads();

    if (t < 64) atomicAdd(&ws[WS_COL + t], scol[t]);
    if (t == 0) atomicAdd(&ws[WS_SUMSQ], ssq);
}

// sum(L2) = 2N*sumsq - 2*||colsum||^2 ; bw = sum(L2)/(N^2-N)/4 ; store 1/(16*bw)
__global__ void mmd_bandwidth(float* __restrict__ ws) {
    const int lane = threadIdx.x;  // 32 threads
    float c0 = ws[WS_COL + lane];
    float c1 = ws[WS_COL + 32 + lane];
    float csq = c0 * c0 + c1 * c1;
#pragma unroll
    for (int off = 16; off > 0; off >>= 1) csq += __shfl_xor(csq, off);
    if (lane == 0) {
        double sumsq = (double)ws[WS_SUMSQ];
        double sumL2 = 2.0 * (double)NROWS * sumsq - 2.0 * (double)csq;
        double bw = sumL2 / ((double)NROWS * (double)NROWS - (double)NROWS);
        bw *= 0.25;  // / KERNEL_MUL^(KERNEL_NUM//2) = 2^2
        ws[WS_CEXP] = (float)(1.0 / (bw * 16.0));
    }
}

// One wave per 16x16 tile of the 8192x8192 kernel matrix; tj >= ti only.
__global__ void __launch_bounds__(256)
mmd_tiles(const float* __restrict__ src, const float* __restrict__ tgt,
          float* __restrict__ ws) {
    const int lane = threadIdx.x & 31;
    const int wave = threadIdx.x >> 5;
    const int ti = blockIdx.x;
    const int tj = blockIdx.y * 8 + wave;
    if (tj < ti) return;  // wave-uniform: surviving waves keep EXEC all-ones for WMMA

    const int laneLo = lane & 15;
    const int laneHi = lane >> 4;
    const int ibase = ti * 16;
    const int jbase = tj * 16;

    const float* baseA = (ibase < NS) ? (src + (size_t)ibase * DIMS)
                                      : (tgt + (size_t)(ibase - NS) * DIMS);
    const float* baseB = (jbase < NS) ? (src + (size_t)jbase * DIMS)
                                      : (tgt + (size_t)(jbase - NS) * DIMS);

    // A fragment (16x4 f32): lane<16 -> row=lane, K=0..1 ; lane>=16 -> row=lane-16, K=2..3
    // B fragment (4x16 f32): same addressing from the row-major tj block (B = X_j^T)
    const float* pa = baseA + (size_t)laneLo * DIMS + 2 * laneHi;
    const float* pb = baseB + (size_t)laneLo * DIMS + 2 * laneHi;

    v8f acc = {};
#pragma unroll
    for (int k = 0; k < 16; ++k) {  // K = 64 in steps of 4
        v2f a = *(const v2f*)(pa + 4 * k);
        v2f b = *(const v2f*)(pb + 4 * k);
        acc = __builtin_amdgcn_wmma_f32_16x16x4_f32(
            /*neg_a=*/false, a, /*neg_b=*/false, b,
            /*c_mod=*/(short)0, acc, /*reuse_a=*/false, /*reuse_b=*/false);
    }

    // C/D layout: VGPR v, lane<16 -> (M=v, N=lane) ; lane>=16 -> (M=v+8, N=lane-16)
    const float sqj = ws[WS_SQ + jbase + laneLo];
    const float* sqi = ws + WS_SQ + ibase + 8 * laneHi;
    const float cExp = ws[WS_CEXP];

    float wsum = 0.f;
#pragma unroll
    for (int v = 0; v < 8; ++v) {
        float L2 = sqi[v] + sqj - 2.f * acc[v];
        float u = __expf(-L2 * cExp);          // exp(-L2/(16*bw))
        float s = u;
        float p = u * u; s += p;               // u^2  : bw*8
        p *= p;          s += p;               // u^4  : bw*4
        p *= p;          s += p;               // u^8  : bw*2
        p *= p;          s += p;               // u^16 : bw*1
        wsum += s;
    }

    float w = ((ti < HALF_T) == (tj < HALF_T)) ? 1.f : -1.f;  // XX/YY vs XY/YX
    if (tj > ti) w *= 2.f;                                    // symmetry doubling
    wsum *= w;

#pragma unroll
    for (int off = 16; off > 0; off >>= 1) wsum += __shfl_xor(wsum, off);
    if (lane == 0) atomicAdd((double*)(ws + WS_ACC), (double)wsum);
}

__global__ void mmd_final(const float* __restrict__ ws, float* __restrict__ out) {
    if (threadIdx.x == 0) {
        double acc = *(const double*)(ws + WS_ACC);
        out[0] = (float)(acc / ((double)NS * (double)NS));
    }
}

extern "C" void kernel_launch(void* const* d_in, const int* in_sizes, int n_in,
                              void* d_out, int out_size, void* d_ws, size_t ws_size,
                              hipStream_t stream) {
    const float* src = (const float*)d_in[0];
    const float* tgt = (const float*)d_in[1];
    float* ws = (float*)d_ws;
    float* out = (float*)d_out;

    mmd_init<<<1, 128, 0, stream>>>(ws);
    mmd_rowstats<<<NROWS / 256, 256, 0, stream>>>(src, tgt, ws);
    mmd_bandwidth<<<1, 32, 0, stream>>>(ws);
    mmd_tiles<<<dim3(NT, NT / 8), 256, 0, stream>>>(src, tgt, ws);
    mmd_final<<<1, 32, 0, stream>>>(ws, out);
}